// UnpadSoftMax_3848290697287
// MI455X (gfx1250) — compile-verified
//
#include <hip/hip_runtime.h>

// ---------------------------------------------------------------------------
// Segmented softmax over packed attention scores (fp16 in / fp16 out).
// One 256-thread block (8 wave32s) per row. Row length s = EPR*256.
// HBM traffic: 1 read + 1 write per element (roofline ~38us @ 23.3 TB/s).
// Data path: GLOBAL_LOAD_ASYNC_TO_LDS_B128 (512B/wave/instr) -> LDS ->
// registers -> LDS -> GLOBAL_STORE_ASYNC_FROM_LDS_B128.
// ---------------------------------------------------------------------------

#if __has_builtin(__builtin_amdgcn_global_load_async_to_lds_b128) && \
    __has_builtin(__builtin_amdgcn_global_store_async_from_lds_b128)
#define HAVE_ASYNC_LDS 1
#else
#define HAVE_ASYNC_LDS 0
#endif

typedef int v4i __attribute__((vector_size(16)));
typedef __attribute__((address_space(1))) v4i* global_v4i_t;  // HBM pointer
typedef __attribute__((address_space(3))) v4i* lds_v4i_t;     // LDS pointer

typedef __attribute__((ext_vector_type(4))) unsigned int u32x4;

#define LOG2E_F 1.44269504088896340736f

__device__ __forceinline__ float wave_max(float v) {
#pragma unroll
  for (int off = 16; off; off >>= 1) v = fmaxf(v, __shfl_xor(v, off, 32));
  return v;
}

__device__ __forceinline__ float wave_sum(float v) {
#pragma unroll
  for (int off = 16; off; off >>= 1) v += __shfl_xor(v, off, 32);
  return v;
}

__device__ __forceinline__ float fast_exp2(float x) {
#if __has_builtin(__builtin_amdgcn_exp2f)
  return __builtin_amdgcn_exp2f(x);   // v_exp_f32
#else
  return exp2f(x);
#endif
}

__device__ __forceinline__ float fast_rcp(float x) {
#if __has_builtin(__builtin_amdgcn_rcpf)
  return __builtin_amdgcn_rcpf(x);    // v_rcp_f32
#else
  return 1.0f / x;
#endif
}

__device__ __forceinline__ void wait_async0() {
#if __has_builtin(__builtin_amdgcn_s_wait_asynccnt)
  __builtin_amdgcn_s_wait_asynccnt(0);
#else
  asm volatile("s_wait_asynccnt 0x0" ::: "memory");
#endif
}

// One row of length EPR*256. gx/gy point at the row base (512B aligned:
// every segment start and every row stride is a multiple of 256 halves).
template <int EPR>
__device__ __forceinline__ void softmax_row(const _Float16* __restrict__ gx,
                                            _Float16* __restrict__ gy,
                                            _Float16* srow, float* red) {
  const int tid  = threadIdx.x;
  const int wave = tid >> 5;
  const int lane = tid & 31;

  // ---- Stage row: HBM -> LDS, 512B chunks (b128 per lane, one chunk/wave)
#if HAVE_ASYNC_LDS
#pragma unroll
  for (int c = 0; c < EPR; ++c) {
    if ((c & 7) == wave) {
      const int boff = c * 512 + lane * 16;
      __builtin_amdgcn_global_load_async_to_lds_b128(
          (global_v4i_t)((char*)gx + boff),
          (lds_v4i_t)((char*)srow + boff), 0, 0);
    }
  }
  wait_async0();
#else
  {
    const u32x4* s = (const u32x4*)gx;
    u32x4*       d = (u32x4*)srow;
    for (int i = tid; i < EPR * 32; i += 256) d[i] = s[i];
  }
#endif
  __syncthreads();

  // ---- Pass A: load EPR values/thread from LDS, row max -------------------
  float v[EPR];
  float m = -__builtin_huge_valf();
#pragma unroll
  for (int k = 0; k < EPR; ++k) {
    v[k] = (float)srow[tid + k * 256];
    m = fmaxf(m, v[k]);
  }
  m = wave_max(m);
  if (lane == 0) red[wave] = m;
  __syncthreads();
  float bm = red[0];
#pragma unroll
  for (int j = 1; j < 8; ++j) bm = fmaxf(bm, red[j]);
  __syncthreads();  // protect red[] before reuse for the sum

  // ---- Pass B: exp once per element (kept in registers), row sum ----------
  float sum = 0.0f;
#pragma unroll
  for (int k = 0; k < EPR; ++k) {
    const float e = fast_exp2((v[k] - bm) * LOG2E_F);
    v[k] = e;
    sum += e;
  }
  sum = wave_sum(sum);
  if (lane == 0) red[wave] = sum;
  __syncthreads();
  float bs = red[0];
#pragma unroll
  for (int j = 1; j < 8; ++j) bs += red[j];
  const float rinv = fast_rcp(bs);

  // ---- Pass C: normalize, fp16 results -> LDS -> HBM (async b128) ---------
#if HAVE_ASYNC_LDS
#pragma unroll
  for (int k = 0; k < EPR; ++k)
    srow[tid + k * 256] = (_Float16)(v[k] * rinv);
  __syncthreads();
#pragma unroll
  for (int c = 0; c < EPR; ++c) {
    if ((c & 7) == wave) {
      const int boff = c * 512 + lane * 16;
      __builtin_amdgcn_global_store_async_from_lds_b128(
          (global_v4i_t)((char*)gy + boff),
          (lds_v4i_t)((char*)srow + boff), 0, 0);
    }
  }
  // S_ENDPGM performs an implicit wait-idle, draining ASYNCcnt.
#else
#pragma unroll
  for (int k = 0; k < EPR; ++k)
    gy[tid + k * 256] = (_Float16)(v[k] * rinv);
#endif
}

// Static layout: SEQ = {2048,1536,1024,512,1792,1280,768,2304}, HEADS = 12.
// Rows per segment = 12*s; cumulative row starts and element starts below.
__global__ __launch_bounds__(256) void UnpadSoftMax_3848290697287_kernel(
    const _Float16* __restrict__ x, _Float16* __restrict__ y) {
  __shared__ _Float16 srow[2304] __attribute__((aligned(16)));
  __shared__ float red[8];

  const int b = (int)blockIdx.x;  // global row id, 0..135167

  int seg;
  if      (b <  24576) seg = 0;
  else if (b <  43008) seg = 1;
  else if (b <  55296) seg = 2;
  else if (b <  61440) seg = 3;
  else if (b <  82944) seg = 4;
  else if (b <  98304) seg = 5;
  else if (b < 107520) seg = 6;
  else                 seg = 7;

  switch (seg) {
    case 0: { const long long base =         0LL + (long long)(b -      0) * 2048;
              softmax_row<8>(x + base, y + base, srow, red); } break;
    case 1: { const long long base =  50331648LL + (long long)(b -  24576) * 1536;
              softmax_row<6>(x + base, y + base, srow, red); } break;
    case 2: { const long long base =  78643200LL + (long long)(b -  43008) * 1024;
              softmax_row<4>(x + base, y + base, srow, red); } break;
    case 3: { const long long base =  91226112LL + (long long)(b -  55296) *  512;
              softmax_row<2>(x + base, y + base, srow, red); } break;
    case 4: { const long long base =  94371840LL + (long long)(b -  61440) * 1792;
              softmax_row<7>(x + base, y + base, srow, red); } break;
    case 5: { const long long base = 132907008LL + (long long)(b -  82944) * 1280;
              softmax_row<5>(x + base, y + base, srow, red); } break;
    case 6: { const long long base = 152567808LL + (long long)(b -  98304) *  768;
              softmax_row<3>(x + base, y + base, srow, red); } break;
    default:{ const long long base = 159645696LL + (long long)(b - 107520) * 2304;
              softmax_row<9>(x + base, y + base, srow, red); } break;
  }
}

extern "C" void kernel_launch(void* const* d_in, const int* in_sizes, int n_in,
                              void* d_out, int out_size, void* d_ws, size_t ws_size,
                              hipStream_t stream) {
  (void)in_sizes; (void)n_in; (void)out_size; (void)d_ws; (void)ws_size;
  const _Float16* x = (const _Float16*)d_in[0];  // fp16 scores, flat
  _Float16*       y = (_Float16*)d_out;          // fp16 softmax, flat
  // 135,168 rows total (sum over segments of HEADS * s_i), one block per row.
  dim3 grid(135168), block(256);
  hipLaunchKernelGGL(UnpadSoftMax_3848290697287_kernel, grid, block, 0, stream,
                     x, y);
}